// RNN_test_54228257080102
// MI455X (gfx1250) — compile-verified
//
#include <hip/hip_runtime.h>

typedef __attribute__((ext_vector_type(2))) float v2f;
typedef __attribute__((ext_vector_type(8))) float v8f;

#define T_STEPS 2048
#define D_IN    32
#define N_HID   100
#define N_PAD   112
#define O_OUT   3
#define KIN     132              // D_IN + N_HID
#define W1_STRIDE 138            // floats; 132 data + pad (even -> aligned b64, conflict-free)
#define BUF_STRIDE 148           // floats; 32 x + 112 spike + 4 pad
#define LDS_W2T_OFF  (N_PAD * W1_STRIDE)          // 15456
#define LDS_BUF0_OFF (LDS_W2T_OFF + N_PAD * 16)   // 17248
#define LDS_BUF1_OFF (LDS_BUF0_OFF + 16 * BUF_STRIDE) // 19616
#define LDS_FLOATS   (LDS_BUF1_OFF + 16 * BUF_STRIDE) // 21984 floats = 87936 B

extern "C" __global__ __launch_bounds__(256)
void dhsnn_fwd_kernel(const float* __restrict__ x,
                      const float* __restrict__ W1,
                      const float* __restrict__ b1,
                      const float* __restrict__ tau1,
                      const float* __restrict__ W2,
                      const float* __restrict__ b2,
                      const float* __restrict__ tau2,
                      const float* __restrict__ mem1_0,
                      const float* __restrict__ mem2_0,
                      float* __restrict__ out)
{
    extern __shared__ float lds[];
    float* w1s  = lds;                 // [112][138]  B-matrix source, row n = W1[n][k]
    float* w2t  = lds + LDS_W2T_OFF;   // [112][16]   W2 transposed+padded
    float* buf0 = lds + LDS_BUF0_OFF;  // [16][148]   k_in ping
    float* buf1 = lds + LDS_BUF1_OFF;  // [16][148]   k_in pong

    const int tid  = threadIdx.x;
    const int wave = tid >> 5;
    const int lane = tid & 31;
    const int ln   = lane & 15;        // A row / B col / D col within tile
    const int hi   = lane >> 4;
    const int mhi  = hi * 8;           // D row offset for this lane half
    const int koff = hi * 2;           // A/B K offset for this lane half
    const int batch0 = blockIdx.x * 16;

    // ---------- setup: W1 -> LDS (zero-padded to 112x138) ----------
    for (int i = tid; i < N_PAD * W1_STRIDE; i += 256) {
        int r = i / W1_STRIDE, c = i - r * W1_STRIDE;
        float v = 0.0f;
        if (r < N_HID && c < KIN) v = W1[r * KIN + c];
        w1s[i] = v;
    }
    // W2^T -> LDS: w2t[n*16+o] = W2[o][n], zero-padded
    for (int i = tid; i < N_PAD * 16; i += 256) {
        int n = i >> 4, o = i & 15;
        float v = 0.0f;
        if (n < N_HID && o < O_OUT) v = W2[o * N_HID + n];
        w2t[i] = v;
    }
    // prime buf0: x_0 in cols 0..31, spikes (cols 32..147) = 0
    for (int i = tid; i < 16 * BUF_STRIDE; i += 256) {
        int r = i / BUF_STRIDE, c = i - r * BUF_STRIDE;
        float v = 0.0f;
        if (c < D_IN) v = x[(size_t)(batch0 + r) * (T_STEPS * D_IN) + c];
        buf0[i] = v;
    }

    // ---------- persistent per-wave register state ----------
    v8f mem1c = {}; v8f spk = {};  float a1v = 0.0f, b1v = 0.0f;
    v8f mem2c = {}; v8f acc = {};  float a2v = 0.0f, b2v = 0.0f;

    if (wave < 7) {
        const int n_abs = wave * 16 + ln;
        if (n_abs < N_HID) {
            a1v = 1.0f / (1.0f + __expf(-tau1[n_abs]));
            b1v = b1[n_abs];
#pragma unroll
            for (int i = 0; i < 8; ++i)
                mem1c[i] = mem1_0[(size_t)(batch0 + mhi + i) * N_HID + n_abs];
        }
    } else {
        if (ln < O_OUT) {
            a2v = 1.0f / (1.0f + __expf(-tau2[ln]));
            b2v = b2[ln];
#pragma unroll
            for (int i = 0; i < 8; ++i)
                mem2c[i] = mem2_0[(size_t)(batch0 + mhi + i) * O_OUT + ln];
        }
    }

    // ---------- time loop (strictly sequential recurrence) ----------
    for (int t = 0; t < T_STEPS; ++t) {
        float* cur = (t & 1) ? buf1 : buf0;
        float* nxt = (t & 1) ? buf0 : buf1;

        __syncthreads();   // B1: prev step's spikes + x_t visible in `cur`

        if (wave < 7) {
            const int    n_abs = wave * 16 + ln;
            const float* arow  = cur + ln * BUF_STRIDE + koff;      // A[m=ln][k]
            const float* brow  = w1s + n_abs * W1_STRIDE + koff;    // B[k][n=ln]
            v8f c0 = {}, c1 = {};
            // K = 132 -> 33 chunks of 4; two accumulator chains to pipeline
#pragma unroll
            for (int k = 0; k < KIN; k += 8) {
                v2f a0 = *(const v2f*)(arow + k);
                v2f bb0 = *(const v2f*)(brow + k);
                c0 = __builtin_amdgcn_wmma_f32_16x16x4_f32(false, a0, false, bb0,
                                                           (short)0, c0, false, false);
                if (k + 4 < KIN) {
                    v2f a1 = *(const v2f*)(arow + k + 4);
                    v2f bb1 = *(const v2f*)(brow + k + 4);
                    c1 = __builtin_amdgcn_wmma_f32_16x16x4_f32(false, a1, false, bb1,
                                                               (short)0, c1, false, false);
                }
            }
            v8f d1 = c0 + c1;
            // LIF update + spike; write spike_{t} into `nxt` for everyone
            float* sdst = nxt + mhi * BUF_STRIDE + (D_IN + n_abs);
#pragma unroll
            for (int i = 0; i < 8; ++i) {
                float m = mem1c[i] * a1v + (1.0f - a1v) * (d1[i] + b1v) - spk[i];
                float s = (m > 1.0f) ? 1.0f : 0.0f;   // Heaviside(mem - VTH)
                mem1c[i] = m;
                spk[i]   = s;
                sdst[i * BUF_STRIDE] = s;
            }
        } else {
            // wave 7: prefetch x_{t+1} into `nxt` cols 0..31 (overlaps layer-1 WMMAs)
            if (t + 1 < T_STEPS) {
                int m  = lane >> 1;
                int cb = (lane & 1) * 16;
                const float4* gx = (const float4*)(x + (size_t)(batch0 + m) * (T_STEPS * D_IN)
                                                     + (size_t)(t + 1) * D_IN + cb);
                float4* lx = (float4*)(nxt + m * BUF_STRIDE + cb);
                lx[0] = gx[0]; lx[1] = gx[1]; lx[2] = gx[2]; lx[3] = gx[3];
            }
        }

        __syncthreads();   // B2: spikes of step t ready in `nxt`

        if (wave == 7) {
            // readout GEMM: d2 = spike_t @ W2^T  (K = 100 -> 25 chunks)
            const float* arow = nxt + ln * BUF_STRIDE + D_IN + koff;
            v8f c2 = {};
#pragma unroll
            for (int k = 0; k < N_HID; k += 4) {
                v2f a = *(const v2f*)(arow + k);
                v2f b;
                b.x = w2t[(k + koff) * 16 + ln];
                b.y = w2t[(k + koff + 1) * 16 + ln];
                c2 = __builtin_amdgcn_wmma_f32_16x16x4_f32(false, a, false, b,
                                                           (short)0, c2, false, false);
            }
#pragma unroll
            for (int i = 0; i < 8; ++i) {
                float m2 = mem2c[i] * a2v + (1.0f - a2v) * (c2[i] + b2v);
                mem2c[i] = m2;
                if (t > 0) acc[i] += m2;   // reference sums mem2_seq[1:]
            }
        }
    }

    // ---------- output: mean over time ----------
    if (wave == 7 && ln < O_OUT) {
#pragma unroll
        for (int i = 0; i < 8; ++i)
            out[(size_t)(batch0 + mhi + i) * O_OUT + ln] = acc[i] * (1.0f / (float)T_STEPS);
    }
}

extern "C" void kernel_launch(void* const* d_in, const int* in_sizes, int n_in,
                              void* d_out, int out_size, void* d_ws, size_t ws_size,
                              hipStream_t stream) {
    const float* x      = (const float*)d_in[0];
    const float* W1     = (const float*)d_in[1];
    const float* b1     = (const float*)d_in[2];
    const float* tau1   = (const float*)d_in[3];
    const float* W2     = (const float*)d_in[4];
    const float* b2     = (const float*)d_in[5];
    const float* tau2   = (const float*)d_in[6];
    const float* mem1_0 = (const float*)d_in[7];
    const float* mem2_0 = (const float*)d_in[8];
    float* out = (float*)d_out;

    const size_t lds_bytes = (size_t)LDS_FLOATS * sizeof(float); // 87936 B < 320 KB/WGP
    dhsnn_fwd_kernel<<<dim3(512 / 16), dim3(256), lds_bytes, stream>>>(
        x, W1, b1, tau1, W2, b2, tau2, mem1_0, mem2_0, out);
}